// VectorQuantizer_38938173506079
// MI455X (gfx1250) — compile-verified
//
#include <hip/hip_runtime.h>

typedef __attribute__((ext_vector_type(2))) float v2f;
typedef __attribute__((ext_vector_type(8))) float v8f;

#define VQ_N 524288
#define VQ_D 64
#define VQ_K 512
#define ROWS_PER_BLOCK 128   // 8 waves * 16 rows
#define NBLOCKS (VQ_N / ROWS_PER_BLOCK)   // 4096
#define FRAG_FLOATS (32 * 16 * 32 * 2)    // 32768 floats = 128 KB (B fragments)
#define SMEM_FLOATS (FRAG_FLOATS + VQ_K + 8)

__global__ __launch_bounds__(256) void vq_main(const float* __restrict__ W,
                                               const float* __restrict__ CB,
                                               float* __restrict__ out,
                                               float* __restrict__ partial) {
  extern __shared__ float smem[];
  float* frag  = smem;                     // WMMA-B fragments, scrambled layout
  float* esq   = smem + FRAG_FLOATS;       // codebook squared norms
  float* wpart = smem + FRAG_FLOATS + VQ_K;

  const int tid  = threadIdx.x;
  const int lane = tid & 31;
  const int wave = tid >> 5;
  const int lo   = lane & 15;
  const int hi   = lane >> 4;

  // ---- Stage codebook into LDS in B-fragment order ----
  // B 4x16 f32 layout (mirrors documented A 16x4): v0 = K{0 | 2}, v1 = K{1 | 3},
  // lanes 0-15 = low half (codes N=0..15 of tile), lanes 16-31 = high half.
  for (int e = tid; e < VQ_K * VQ_D; e += 256) {
    int c = e >> 6, d = e & 63;
    int j = c >> 4, l = c & 15;       // column tile, code-in-tile
    int s = d >> 2, off = d & 3;      // k-step, dim-in-step
    int fl = l + 16 * (off >> 1);     // fragment lane
    int v  = off & 1;                 // which of the 2 VGPRs
    frag[((((j * 16 + s) * 32) + fl) << 1) | v] = CB[e];
  }
  for (int c = tid; c < VQ_K; c += 256) {
    const float* cp = CB + c * VQ_D;
    float acc = 0.f;
#pragma unroll
    for (int d = 0; d < VQ_D; ++d) acc += cp[d] * cp[d];
    esq[c] = acc;
  }
  __syncthreads();

  // ---- Load A fragments: this wave's 16 rows, entirely in registers ----
  // A 16x4 f32 layout: lane%16 = M, v0 = K=4s+2*hi, v1 = K=4s+2*hi+1.
  const long rowBase = (long)blockIdx.x * ROWS_PER_BLOCK + (long)wave * 16;
  const float* wrow = W + (rowBase + lo) * VQ_D;
  v2f a[16];
#pragma unroll
  for (int s = 0; s < 16; ++s)
    a[s] = *(const v2f*)(wrow + 4 * s + 2 * hi);

  float bestv[8];
  int   besti[8];
#pragma unroll
  for (int r = 0; r < 8; ++r) { bestv[r] = 3.402823466e38f; besti[r] = 0; }

  // ---- Sweep all 32 column tiles of the codebook ----
  for (int j = 0; j < 32; ++j) {
    v8f acc = {};
    const float* fb = frag + ((j * 16) * 32 + lane) * 2;
#pragma unroll
    for (int s = 0; s < 16; ++s) {
      v2f b = *(const v2f*)(fb + s * 64);
      acc = __builtin_amdgcn_wmma_f32_16x16x4_f32(
          false, a[s], false, b, (short)0, acc, false, false);
    }
    // C/D layout: VGPR r, lanes 0-15 -> (row=r, col=lo); lanes 16-31 -> (row=r+8, col=lo)
    float eq = esq[j * 16 + lo];
    int   ix = j * 16 + lo;
#pragma unroll
    for (int r = 0; r < 8; ++r) {
      float val = eq - 2.0f * acc[r];   // x_sq is row-constant: dropped
      if (val < bestv[r]) { bestv[r] = val; besti[r] = ix; }
    }
  }

  // ---- Cross-lane argmin within each 16-lane half (tie -> lowest index) ----
#pragma unroll
  for (int r = 0; r < 8; ++r) {
    float v = bestv[r];
    int   i = besti[r];
#pragma unroll
    for (int m = 1; m <= 8; m <<= 1) {
      float ov = __shfl_xor(v, m, 32);
      int   oi = __shfl_xor(i, m, 32);
      if (ov < v || (ov == v && oi < i)) { v = ov; i = oi; }
    }
    besti[r] = i;
  }

  // ---- Gather, write straight-through output, accumulate loss ----
  float lsum = 0.f;
#pragma unroll
  for (int r = 0; r < 8; ++r) {
    int i0 = __shfl(besti[r], 0, 32);    // consensus of lanes 0-15: row r
    int i1 = __shfl(besti[r], 16, 32);   // consensus of lanes 16-31: row r+8
#pragma unroll
    for (int half = 0; half < 2; ++half) {
      long row = rowBase + r + 8 * half;
      int  idx = half ? i1 : i0;
      const float* q = CB + (long)idx * VQ_D + 2 * lane;
      const float* w = W + row * VQ_D + 2 * lane;
      float q0 = q[0], q1 = q[1];
      float w0 = w[0], w1 = w[1];
      float d0 = q0 - w0, d1 = q1 - w1;
      lsum += d0 * d0 + d1 * d1;
      float* o = out + row * VQ_D + 2 * lane;
      o[0] = w0 + d0;                    // w + sg(q - w), reference rounding
      o[1] = w1 + d1;
    }
  }

  // deterministic reductions: wave shuffle tree -> LDS -> fixed-order sum
#pragma unroll
  for (int m = 1; m <= 16; m <<= 1) lsum += __shfl_xor(lsum, m, 32);
  if (lane == 0) wpart[wave] = lsum;
  __syncthreads();
  if (tid == 0) {
    float s = 0.f;
    for (int v = 0; v < 8; ++v) s += wpart[v];
    partial[blockIdx.x] = s;
  }
}

__global__ __launch_bounds__(256) void vq_loss_reduce(const float* __restrict__ partial,
                                                      float* __restrict__ out_loss) {
  __shared__ float sm[256];
  int t = threadIdx.x;
  float s = 0.f;
  for (int i = t; i < NBLOCKS; i += 256) s += partial[i];
  sm[t] = s;
  __syncthreads();
  for (int k = 128; k > 0; k >>= 1) {
    if (t < k) sm[t] += sm[t + k];
    __syncthreads();
  }
  if (t == 0)
    out_loss[0] = sm[0] * (1.1f / ((float)VQ_N * (float)VQ_D));  // mean + 0.1*mean
}

extern "C" void kernel_launch(void* const* d_in, const int* in_sizes, int n_in,
                              void* d_out, int out_size, void* d_ws, size_t ws_size,
                              hipStream_t stream) {
  const float* W  = (const float*)d_in[0];   // [N, D] fp32
  const float* CB = (const float*)d_in[1];   // [K, D] fp32
  float* out      = (float*)d_out;           // [N*D] quantized_st, then [1] vq_loss
  float* partial  = (float*)d_ws;            // NBLOCKS fp32 partial sums

  size_t smem = SMEM_FLOATS * sizeof(float); // ~130 KB of 320 KB/WGP
  vq_main<<<NBLOCKS, 256, smem, stream>>>(W, CB, out, partial);
  vq_loss_reduce<<<1, 256, 0, stream>>>(partial, out + (size_t)VQ_N * VQ_D);
  (void)in_sizes; (void)n_in; (void)out_size; (void)ws_size;
}